// MemoryModule_17145509445673
// MI455X (gfx1250) — compile-verified
//
#include <hip/hip_runtime.h>
#include <stdint.h>

#define DEP    5
#define TOTAL  4096
#define SLOT   5120
#define NSEG   16
#define CH     64                // s-floats per staged chunk
#define NCH    (SLOT / CH)       // 80 chunks
#define PITCH  68                // padded LDS row pitch (dwords) -> conflict-free b64 reads
#define MAXTILES 272             // sum ceil(cnt_b/16) <= 256 + 15

typedef float v2f __attribute__((ext_vector_type(2)));
typedef float v8f __attribute__((ext_vector_type(8)));

// ---------------------------------------------------------------- setup ----
__global__ __launch_bounds__(256) void mm_setup_kernel(
    const float* __restrict__ emb, const float* __restrict__ freq,
    const int* __restrict__ ids,
    float* __restrict__ f_emb, int* __restrict__ seg_start, int* __restrict__ seg_cnt,
    int* __restrict__ tile_b, int* __restrict__ tile_t0, int* __restrict__ tile_end,
    int* __restrict__ n_tiles)
{
  const int tid = threadIdx.x;
  for (int i = tid; i < TOTAL * DEP; i += 256) {
    const int t = i / DEP;
    f_emb[i] = emb[i] * freq[t];
  }
  if (tid < NSEG) {              // ids are sorted; find [start,count) per segment
    int st = 0, cnt = 0;
    for (int t = 0; t < TOTAL; ++t) {
      if (ids[t] == tid) { if (cnt == 0) st = t; ++cnt; }
    }
    seg_start[tid] = st;
    seg_cnt[tid]  = cnt;
  }
  __syncthreads();
  if (tid == 0) {                // tiles of 16 rows, never crossing a segment
    int nt = 0;
    for (int b = 0; b < NSEG; ++b) {
      const int st = seg_start[b], en = st + seg_cnt[b];
      for (int t0 = st; t0 < en; t0 += 16) {
        tile_b[nt] = b; tile_t0[nt] = t0; tile_end[nt] = en; ++nt;
      }
    }
    *n_tiles = nt;
  }
}

// ------------------------------------------------- write pass (segment sum) ----
// grid = (SLOT/256, NSEG), block = 64; each thread owns a float4 of s.
__global__ __launch_bounds__(64) void mm_write_kernel(
    const float* __restrict__ addr, const float* __restrict__ memmat,
    const float* __restrict__ f_emb, const int* __restrict__ seg_start,
    const int* __restrict__ seg_cnt, float* __restrict__ mem_ws)
{
  const int b = blockIdx.y;
  const int s = blockIdx.x * 256 + threadIdx.x * 4;
  float acc[DEP][4] = {};
  const int t0 = seg_start[b], cnt = seg_cnt[b];
  for (int i = 0; i < cnt; ++i) {
    const int t = t0 + i;
#pragma unroll
    for (int d = 0; d < DEP; ++d) {
      const float4 av = *(const float4*)(addr + ((size_t)d * TOTAL + t) * SLOT + s);
      const float  f  = f_emb[t * DEP + d];
      acc[d][0] += av.x * f; acc[d][1] += av.y * f;
      acc[d][2] += av.z * f; acc[d][3] += av.w * f;
    }
  }
#pragma unroll
  for (int d = 0; d < DEP; ++d) {
    const size_t o = ((size_t)b * DEP + d) * SLOT + s;
    const float4 mv = *(const float4*)(memmat + o);
    float4 r;
    r.x = mv.x + acc[d][0]; r.y = mv.y + acc[d][1];
    r.z = mv.z + acc[d][2]; r.w = mv.w + acc[d][3];
    *(float4*)(mem_ws + o) = r;
  }
}

// ------------------------------------ read pass: async-LDS pipeline + WMMA ----
// block = 160 (5 waves, wave w == d); one block per 16-row tile.
__global__ __launch_bounds__(160) void mm_read_kernel(
    const float* __restrict__ addr, const float* __restrict__ emb,
    const float* __restrict__ mem_ws, const int* __restrict__ tile_b,
    const int* __restrict__ tile_t0, const int* __restrict__ tile_end,
    const int* __restrict__ n_tiles,
    float* __restrict__ out_head, float* __restrict__ out_info,
    float* __restrict__ out_br)
{
  __shared__ __align__(16) float sA[DEP][2][16 * PITCH]; // per-wave double-buffered A tile
  __shared__ __align__(16) float sB[DEP][2][PITCH];      // per-wave double-buffered mem chunk
  __shared__ float sred[16][DEP];

  const int bid = blockIdx.x;
  if (bid >= *n_tiles) return;

  const int w    = threadIdx.x >> 5;   // wave id == d
  const int lane = threadIdx.x & 31;
  const int d    = w;
  const int b    = tile_b[bid];
  const int t0   = tile_t0[bid];
  const int tend = tile_end[bid];

  const float* abase = addr  + (size_t)d * TOTAL * SLOT;
  const float* mbase = mem_ws + ((size_t)b * DEP + d) * SLOT;

  // Stage one 16x64 addr tile + 64-float mem chunk into this wave's LDS buffers
  // with CDNA5 async-to-LDS loads (tracked on ASYNCcnt): 9 async instrs / chunk.
  auto stage = [&](int k, int buf) {
    const int s0 = k * CH;
#pragma unroll
    for (int it = 0; it < 8; ++it) {
      const int q   = lane + it * 32;        // 256 b128-quads: 16 rows x 16 quads
      const int row = q >> 4, c4 = q & 15;
      int t = t0 + row; if (t >= tend) t = tend - 1;   // clamp tail rows (unused results)
      const unsigned long long g =
          (unsigned long long)(uintptr_t)(abase + (size_t)t * SLOT + s0 + c4 * 4);
      const unsigned loff = (unsigned)(uintptr_t)&sA[w][buf][row * PITCH + c4 * 4];
      asm volatile("global_load_async_to_lds_b128 %0, %1, off"
                   :: "v"(loff), "v"(g) : "memory");
    }
    if (lane < 16) {
      const unsigned long long g =
          (unsigned long long)(uintptr_t)(mbase + s0 + lane * 4);
      const unsigned loff = (unsigned)(uintptr_t)&sB[w][buf][lane * 4];
      asm volatile("global_load_async_to_lds_b128 %0, %1, off"
                   :: "v"(loff), "v"(g) : "memory");
    }
  };

  v8f c = {0, 0, 0, 0, 0, 0, 0, 0};
  const int kh   = lane >> 4;    // K-half select (ISA 16x4 f32 A layout)
  const int rowm = lane & 15;

  stage(0, 0);
  for (int k = 0; k < NCH; ++k) {
    if (k + 1 < NCH) {
      stage(k + 1, (k + 1) & 1);
      asm volatile("s_wait_asynccnt 9" ::: "memory");  // prev chunk's 9 async ops done
    } else {
      asm volatile("s_wait_asynccnt 0" ::: "memory");
    }
    const float* pA = &sA[w][k & 1][rowm * PITCH + kh * 2];
    const float* pB = &sB[w][k & 1][kh * 2];
#pragma unroll
    for (int step = 0; step < 16; ++step) {
      // A[16x4] f32: lane<16 -> (M=lane, K=0,1), lane>=16 -> (M=lane-16, K=2,3)
      const v2f a  = *(const v2f*)(pA + step * 4);
      // B[4x16] f32 with mem value replicated over all N columns: same b64 pattern
      const v2f bv = *(const v2f*)(pB + step * 4);
      c = __builtin_amdgcn_wmma_f32_16x16x4_f32(
              false, a, false, bv, (short)0, c, false, false);
    }
  }

  // D column 0: lane 0 holds rows 0..7 in c[0..7], lane 16 holds rows 8..15.
  if (lane == 0 || lane == 16) {
    const int rb = (lane == 16) ? 8 : 0;
#pragma unroll
    for (int r = 0; r < 8; ++r) {
      const int t = t0 + rb + r;
      if (t < tend) {
        const float br = c[r];
        out_br[t * DEP + d] = br;
        const float ci = br / emb[t * DEP + d];
        out_info[t * DEP + d] = ci;
        sred[rb + r][d] = ci;
      }
    }
  }
  __syncthreads();
  if (threadIdx.x < 16) {
    const int t = t0 + threadIdx.x;
    if (t < tend) {
      float m = sred[threadIdx.x][0];
#pragma unroll
      for (int dd = 1; dd < DEP; ++dd) m = fminf(m, sred[threadIdx.x][dd]);
      out_head[t] = m;
    }
  }
}

// ------------------------------------------------------------------ launch ----
extern "C" void kernel_launch(void* const* d_in, const int* in_sizes, int n_in,
                              void* d_out, int out_size, void* d_ws, size_t ws_size,
                              hipStream_t stream)
{
  (void)in_sizes; (void)n_in; (void)out_size; (void)ws_size;
  const float* addr   = (const float*)d_in[0];  // [5,4096,5120]
  const float* emb    = (const float*)d_in[1];  // [4096,5]
  const float* freq   = (const float*)d_in[2];  // [4096]
  const float* memmat = (const float*)d_in[3];  // [16,5,5120]
  const int*   ids    = (const int*)d_in[4];    // [4096] sorted

  char* ws = (char*)d_ws;
  float* f_emb  = (float*)ws;                                   // T*D floats
  float* mem_ws = (float*)(ws + (size_t)TOTAL * DEP * 4);       // B*D*S floats
  int* ip = (int*)(ws + (size_t)TOTAL * DEP * 4 + (size_t)NSEG * DEP * SLOT * 4);
  int* seg_start = ip;
  int* seg_cnt   = ip + NSEG;
  int* tile_b    = ip + 2 * NSEG;
  int* tile_t0   = tile_b + MAXTILES;
  int* tile_end  = tile_t0 + MAXTILES;
  int* n_tiles   = tile_end + MAXTILES;

  float* out_head = (float*)d_out;              // [T]
  float* out_info = out_head + TOTAL;           // [T,D]
  float* out_br   = out_info + TOTAL * DEP;     // [T,D]

  hipLaunchKernelGGL(mm_setup_kernel, dim3(1), dim3(256), 0, stream,
                     emb, freq, ids, f_emb, seg_start, seg_cnt,
                     tile_b, tile_t0, tile_end, n_tiles);
  hipLaunchKernelGGL(mm_write_kernel, dim3(SLOT / 256, NSEG), dim3(64), 0, stream,
                     addr, memmat, f_emb, seg_start, seg_cnt, mem_ws);
  hipLaunchKernelGGL(mm_read_kernel, dim3(MAXTILES), dim3(160), 0, stream,
                     addr, emb, mem_ws, tile_b, tile_t0, tile_end, n_tiles,
                     out_head, out_info, out_br);
}